// AttentiveGRU1_3891240370403
// MI455X (gfx1250) — compile-verified
//
#include <hip/hip_runtime.h>
#include <math.h>

typedef __attribute__((ext_vector_type(2))) float v2f;
typedef __attribute__((ext_vector_type(8))) float v8f;

#define DD 128

// ---------- float atomic max via int/uint trick (init must be -inf) ----------
__device__ __forceinline__ void atomicMaxF32(float* addr, float val) {
  if (val >= 0.0f) {
    atomicMax((int*)addr, __float_as_int(val));
  } else {
    atomicMin((unsigned int*)addr, __float_as_uint(val));
  }
}

// ---------- init: m = -inf, s & agg = 0 (must run every launch) ----------
__global__ void init_kernel(float* __restrict__ m3, float* __restrict__ zbase,
                            long ninf, long nzero) {
  long i = (long)blockIdx.x * blockDim.x + threadIdx.x;
  if (i < ninf) m3[i] = -INFINITY;
  if (i < nzero) zbase[i] = 0.0f;
}

// ---------- segment max of 3 logit channels by dst ----------
__global__ void seg_max_kernel(const int* __restrict__ dst,
                               const float* __restrict__ l0, const float* __restrict__ l1,
                               const float* __restrict__ l2,
                               float* __restrict__ m3, int E, int N) {
  int e = blockIdx.x * blockDim.x + threadIdx.x;
  if (e >= E) return;
  int d = dst[e];
  atomicMaxF32(&m3[d],         l0[e]);
  atomicMaxF32(&m3[N + d],     l1[e]);
  atomicMaxF32(&m3[2 * N + d], l2[e]);
}

// ---------- e = exp(logit - m[dst]); store per-edge; segment sum ----------
__global__ void exp_sum_kernel(const int* __restrict__ dst,
                               const float* __restrict__ l0, const float* __restrict__ l1,
                               const float* __restrict__ l2,
                               const float* __restrict__ m3,
                               float* __restrict__ s3, float* __restrict__ abuf,
                               int E, int N) {
  int e = blockIdx.x * blockDim.x + threadIdx.x;
  if (e >= E) return;
  int d = dst[e];
  float v0 = expf(l0[e] - m3[d]);
  float v1 = expf(l1[e] - m3[N + d]);
  float v2 = expf(l2[e] - m3[2 * N + d]);
  abuf[e] = v0;
  abuf[(size_t)E + e] = v1;
  abuf[(size_t)2 * E + e] = v2;
  atomicAdd(&s3[d], v0);
  atomicAdd(&s3[N + d], v1);
  atomicAdd(&s3[2 * N + d], v2);
}

// Stage a 128xNcolsTile B chunk into LDS with K-pair swizzle:
// element (k, col) -> Bsh[(k>>1)*256 + col*2 + (k&1)]
// so a WMMA B fragment {k0, k0+1} x col is one aligned 64-bit LDS load.
__device__ __forceinline__ void stage_B_swizzled(float* Bsh, const float* B,
                                                 size_t rowStride, int tid) {
  for (int i = tid; i < 2048; i += 256) {
    int q = i >> 5;             // k pair index 0..63
    int c = (i & 31) * 4;       // col group
    float4 r0 = *(const float4*)&B[(size_t)(2 * q) * rowStride + c];
    float4 r1 = *(const float4*)&B[(size_t)(2 * q + 1) * rowStride + c];
    float2* dp = (float2*)&Bsh[q * 256 + c * 2];
    dp[0] = make_float2(r0.x, r1.x);
    dp[1] = make_float2(r0.y, r1.y);
    dp[2] = make_float2(r0.z, r1.z);
    dp[3] = make_float2(r0.w, r1.w);
  }
}

// ---------- edge GEMM + softmax-normalize + scatter-add ----------
// agg[dst[e], :] += (abuf[e]/s[dst[e]]) * (ef[e,:] @ W + b)
// Block: 8 waves * 32 edge rows = 256 rows; W (128x128 f32) swizzled in LDS.
__global__ __launch_bounds__(256, 1)
void edge_msg_wmma(const float* __restrict__ ef, const float* __restrict__ W,
                   const float* __restrict__ bvec, const int* __restrict__ dst,
                   const float* __restrict__ abuf, const float* __restrict__ ssum,
                   float* __restrict__ agg, int E) {
  __shared__ __align__(16) float Wsh[DD * DD];
  int tid = threadIdx.x;
  stage_B_swizzled(Wsh, W, DD, tid);
  __syncthreads();

  int wave = tid >> 5;
  int lane = tid & 31;
  int laneM = lane & 15;        // row within 16-row tile (A) / col within 16 (B,D)
  int khalf = lane >> 4;        // which K pair of the 4-wide K step
  int rowBase = blockIdx.x * 256 + wave * 32;

  __builtin_prefetch(dst + (rowBase < E ? rowBase : 0), 0, 0);

  v8f acc0[8] = {};
  v8f acc1[8] = {};
  int r0 = min(rowBase + laneM, E - 1);
  int r1 = min(rowBase + 16 + laneM, E - 1);
  const float* arow0 = ef + (size_t)r0 * DD + 2 * khalf;
  const float* arow1 = ef + (size_t)r1 * DD + 2 * khalf;
  for (int kk = 0; kk < 32; ++kk) {
    v2f af0 = *(const v2f*)(arow0 + 4 * kk);   // cols 4kk+2*khalf, +1
    v2f af1 = *(const v2f*)(arow1 + 4 * kk);
    const v2f* bbase = (const v2f*)&Wsh[(2 * kk + khalf) * 256];
#pragma unroll
    for (int nt = 0; nt < 8; ++nt) {
      v2f bf = bbase[nt * 16 + laneM];
      acc0[nt] = __builtin_amdgcn_wmma_f32_16x16x4_f32(false, af0, false, bf,
                                                       (short)0, acc0[nt], false, false);
      acc1[nt] = __builtin_amdgcn_wmma_f32_16x16x4_f32(false, af1, false, bf,
                                                       (short)0, acc1[nt], false, false);
    }
  }

  // epilogue: per output row -> edge id, attention weight, scatter-add
  int dn[2][8]; float aw[2][8]; bool ok[2][8];
#pragma unroll
  for (int t = 0; t < 2; ++t) {
#pragma unroll
    for (int v = 0; v < 8; ++v) {
      int e = rowBase + t * 16 + v + 8 * khalf;
      ok[t][v] = (e < E);
      int ec = ok[t][v] ? e : 0;
      int d = dst[ec];
      dn[t][v] = d;
      aw[t][v] = abuf[ec] / ssum[d];
    }
  }
#pragma unroll
  for (int nt = 0; nt < 8; ++nt) {
    int col = nt * 16 + laneM;
    float bb = bvec[col];
#pragma unroll
    for (int v = 0; v < 8; ++v) {
      if (ok[0][v]) {
        atomicAdd(&agg[(size_t)dn[0][v] * DD + col], aw[0][v] * (acc0[nt][v] + bb));
      }
      if (ok[1][v]) {
        atomicAdd(&agg[(size_t)dn[1][v] * DD + col], aw[1][v] * (acc1[nt][v] + bb));
      }
    }
  }
}

// ---------- generic node GEMM: C = act(concat(A0,A1,A2)) @ B + bias ----------
// A chunks are each [nrows,128]; K = 128*kchunks; B is [K, ncols] row-major.
// Block = 256 rows x 128 cols; B chunk swizzled in LDS; optional ELU on A.
__global__ __launch_bounds__(256, 1)
void gemm_wmma(const float* __restrict__ A0, const float* __restrict__ A1,
               const float* __restrict__ A2,
               const float* __restrict__ B, const float* __restrict__ bias,
               float* __restrict__ Cout,
               int nrows, int kchunks, int ncols, int do_elu) {
  __shared__ __align__(16) float Bsh[DD * DD];
  int tid = threadIdx.x;
  int wave = tid >> 5, lane = tid & 31;
  int laneM = lane & 15, khalf = lane >> 4;
  int rowBase = blockIdx.x * 256 + wave * 32;
  int colBase = blockIdx.y * 128;
  int r0 = min(rowBase + laneM, nrows - 1);
  int r1 = min(rowBase + 16 + laneM, nrows - 1);

  v8f acc0[8] = {};
  v8f acc1[8] = {};
  for (int kc = 0; kc < kchunks; ++kc) {
    const float* Asrc = (kc == 0) ? A0 : ((kc == 1) ? A1 : A2);
    if (kc) __syncthreads();
    stage_B_swizzled(Bsh, B + (size_t)(kc * 128) * ncols + colBase, ncols, tid);
    __syncthreads();

    const float* arow0 = Asrc + (size_t)r0 * DD + 2 * khalf;
    const float* arow1 = Asrc + (size_t)r1 * DD + 2 * khalf;
    for (int kk = 0; kk < 32; ++kk) {
      v2f af0 = *(const v2f*)(arow0 + 4 * kk);
      v2f af1 = *(const v2f*)(arow1 + 4 * kk);
      if (do_elu) {
        af0.x = af0.x > 0.0f ? af0.x : expm1f(af0.x);
        af0.y = af0.y > 0.0f ? af0.y : expm1f(af0.y);
        af1.x = af1.x > 0.0f ? af1.x : expm1f(af1.x);
        af1.y = af1.y > 0.0f ? af1.y : expm1f(af1.y);
      }
      const v2f* bbase = (const v2f*)&Bsh[(2 * kk + khalf) * 256];
#pragma unroll
      for (int nt = 0; nt < 8; ++nt) {
        v2f bf = bbase[nt * 16 + laneM];
        acc0[nt] = __builtin_amdgcn_wmma_f32_16x16x4_f32(false, af0, false, bf,
                                                         (short)0, acc0[nt], false, false);
        acc1[nt] = __builtin_amdgcn_wmma_f32_16x16x4_f32(false, af1, false, bf,
                                                         (short)0, acc1[nt], false, false);
      }
    }
  }

#pragma unroll
  for (int nt = 0; nt < 8; ++nt) {
    int col = colBase + nt * 16 + laneM;
    float bb = bias[col];
#pragma unroll
    for (int v = 0; v < 8; ++v) {
      int row0 = rowBase + v + 8 * khalf;
      int row1 = rowBase + 16 + v + 8 * khalf;
      if (row0 < nrows) Cout[(size_t)row0 * ncols + col] = acc0[nt][v] + bb;
      if (row1 < nrows) Cout[(size_t)row1 * ncols + col] = acc1[nt][v] + bb;
    }
  }
}

// ---------- GRU gates + ReLU ----------
__global__ void gru_gate_kernel(const float* __restrict__ gi, const float* __restrict__ gh,
                                const float* __restrict__ h, float* __restrict__ out,
                                long total) {
  long idx = (long)blockIdx.x * blockDim.x + threadIdx.x;
  if (idx >= total) return;
  long row = idx >> 7;
  int col = (int)(idx & 127);
  const float* gir = gi + row * 384;
  const float* ghr = gh + row * 384;
  float r = 1.0f / (1.0f + expf(-(gir[col] + ghr[col])));
  float z = 1.0f / (1.0f + expf(-(gir[128 + col] + ghr[128 + col])));
  float n = tanhf(gir[256 + col] + r * ghr[256 + col]);
  float hv = h[idx];
  float hn = (1.0f - z) * n + z * hv;
  out[idx] = hn > 0.0f ? hn : 0.0f;
}

extern "C" void kernel_launch(void* const* d_in, const int* in_sizes, int n_in,
                              void* d_out, int out_size, void* d_ws, size_t ws_size,
                              hipStream_t stream) {
  (void)n_in; (void)out_size; (void)ws_size;
  const int*   dst  = (const int*)d_in[0];
  const float* l0   = (const float*)d_in[1];
  const float* l1   = (const float*)d_in[2];
  const float* l2   = (const float*)d_in[3];
  const float* ef[3]= {(const float*)d_in[4], (const float*)d_in[5], (const float*)d_in[6]};
  const float* nf[3]= {(const float*)d_in[7], (const float*)d_in[8], (const float*)d_in[9]};
  const float* W[3] = {(const float*)d_in[10], (const float*)d_in[12], (const float*)d_in[14]};
  const float* bW[3]= {(const float*)d_in[11], (const float*)d_in[13], (const float*)d_in[15]};
  const float* Wa   = (const float*)d_in[16];
  const float* ba   = (const float*)d_in[17];
  const float* Wn   = (const float*)d_in[18];
  const float* bn   = (const float*)d_in[19];
  const float* W_ih = (const float*)d_in[20];
  const float* b_ih = (const float*)d_in[21];
  const float* W_hh = (const float*)d_in[22];
  const float* b_hh = (const float*)d_in[23];

  const int E = in_sizes[0];
  const int N = in_sizes[7] / DD;
  const size_t ND = (size_t)N * DD;

  // workspace layout (floats): m3 | s3 | agg(3*N*D) | abuf(3E) | ctx | h | gi | gh
  float* ws   = (float*)d_ws;
  size_t off  = 0;
  float* m3   = ws + off; off += (size_t)3 * N;
  float* s3   = ws + off; off += (size_t)3 * N;        // s3 and agg contiguous -> one zero pass
  float* agg  = ws + off; off += 3 * ND;
  float* abuf = ws + off; off += (size_t)3 * E;
  float* ctx  = ws + off; off += ND;
  float* hbuf = ws + off; off += ND;
  float* gi   = ws + off; off += (size_t)3 * ND;
  float* gh   = ws + off; off += (size_t)3 * ND;

  const long ninf  = (long)3 * N;
  const long nzero = (long)3 * N + (long)3 * ND;
  init_kernel<<<(unsigned)((nzero + 255) / 256), 256, 0, stream>>>(m3, s3, ninf, nzero);

  const unsigned egrid = (unsigned)((E + 255) / 256);
  seg_max_kernel<<<egrid, 256, 0, stream>>>(dst, l0, l1, l2, m3, E, N);
  exp_sum_kernel<<<egrid, 256, 0, stream>>>(dst, l0, l1, l2, m3, s3, abuf, E, N);

  const unsigned etiles = (unsigned)((E + 255) / 256);
  for (int k = 0; k < 3; ++k) {
    edge_msg_wmma<<<etiles, 256, 0, stream>>>(ef[k], W[k], bW[k], dst,
                                              abuf + (size_t)k * E, s3 + (size_t)k * N,
                                              agg + (size_t)k * ND, E);
  }

  dim3 g128((unsigned)((N + 255) / 256), 1);
  dim3 g384((unsigned)((N + 255) / 256), 3);
  // context = elu(agg) @ Wa + ba   ([N,384] @ [384,128])
  gemm_wmma<<<g128, 256, 0, stream>>>(agg, agg + ND, agg + 2 * ND, Wa, ba, ctx,
                                      N, 3, DD, 1);
  // h = [nf1 nf2 nf3] @ Wn + bn
  gemm_wmma<<<g128, 256, 0, stream>>>(nf[0], nf[1], nf[2], Wn, bn, hbuf,
                                      N, 3, DD, 0);
  // gi = context @ W_ih + b_ih     ([N,128] @ [128,384])
  gemm_wmma<<<g384, 256, 0, stream>>>(ctx, ctx, ctx, W_ih, b_ih, gi,
                                      N, 1, 3 * DD, 0);
  // gh = h @ W_hh + b_hh
  gemm_wmma<<<g384, 256, 0, stream>>>(hbuf, hbuf, hbuf, W_hh, b_hh, gh,
                                      N, 1, 3 * DD, 0);

  const long total = (long)N * DD;
  gru_gate_kernel<<<(unsigned)((total + 255) / 256), 256, 0, stream>>>(
      gi, gh, hbuf, (float*)d_out, total);
}